// RandomEigenframeFilter_16801912062653
// MI455X (gfx1250) — compile-verified
//
#include <hip/hip_runtime.h>
#include <hip/hip_bf16.h>

// ---------------------------------------------------------------------------
// RandomEigenframeFilter for MI455X (gfx1250, wave32, WMMA)
//
// Shapes: B=2, F=4 (BF=8), N=20000, Cin=64, Cout=64, KM=256
//   P[bf,i,m]  = sum_n x[bf,n,i] * D[bf,n,m]        (WMMA GEMM, split-K + atomics)
//   St[bf,m,o] = scale * sum_i C[o,i,m] * P[bf,i,m] (tiny VALU kernel, transposed out)
//   out[bf,n,o]= sum_m D[bf,n,m] * St[bf,m,o] + bias[o]  (WMMA GEMM, LDS-staged D)
//
// All big-GEMM loads use wave-uniform SGPR bases (readfirstlane'd wave id) +
// one 32-bit lane offset + instruction immediates -> SADDR-form global loads,
// with only scalar pointer bumps per unroll group.
// ---------------------------------------------------------------------------

typedef __attribute__((ext_vector_type(2))) float v2f;
typedef __attribute__((ext_vector_type(8))) float v8f;

#define BF    8
#define NNODE 20000
#define CIN   64
#define COUT  64
#define KM    256
#define SCALE 0.1767766952966368811f  // 1/sqrt(32)

#define PROJ_CHUNKS 50                // 50 chunks * 100 steps * 4 rows = 20000
#define PROJ_STEPS  100               // divisible by UNROLL=4

// ---------------------------------------------------------------------------
// zero the P accumulator (needed for split-K atomics)
// ---------------------------------------------------------------------------
__global__ void __launch_bounds__(256) zero_kernel(float* __restrict__ p, int n) {
    int i = blockIdx.x * 256 + threadIdx.x;
    if (i < n) p[i] = 0.0f;
}

// ---------------------------------------------------------------------------
// Kernel 1: P[bf,i,m] += sum_{n in chunk} x[bf,n,i] * D[bf,n,m]
// GEMM M=Cin(64) x N=KM(256), K = node dim. Contraction dim is the row dim
// of both x and D -> coalesced loads. 8 waves: 4 i-tiles x 2 m-groups(8 tiles).
// grid = (PROJ_CHUNKS, BF)
// ---------------------------------------------------------------------------
__global__ void __launch_bounds__(256) proj_kernel(const float* __restrict__ x,
                                                   const float* __restrict__ D,
                                                   float* __restrict__ P) {
    const int chunk = blockIdx.x;
    const int bf    = blockIdx.y;
    const int lane  = threadIdx.x & 31;
    const int wave  = __builtin_amdgcn_readfirstlane(threadIdx.x >> 5);
    const int half  = lane >> 4;     // A/B frag: half 0 holds K0/K1, half 1 K2/K3
    const int l16   = lane & 15;
    const int i_tile  = wave & 3;    // 4 i-tiles cover Cin=64
    const int m_group = wave >> 2;   // 2 groups x 8 m-tiles cover KM=256

    const int k_begin = chunk * PROJ_STEPS * 4;

    // wave-uniform SGPR bases; lane-varying part lives in a 32-bit voffset
    const float* __restrict__ xb  = x + (size_t)bf * NNODE * CIN
                                      + (size_t)k_begin * CIN + i_tile * 16;
    const float* __restrict__ Dbp = D + (size_t)bf * NNODE * KM
                                      + (size_t)k_begin * KM + m_group * 128;

    const unsigned xoff = (unsigned)(half * (2 * CIN) + l16);
    const unsigned doff = (unsigned)(half * (2 * KM) + l16);

    v8f acc[8];
#pragma unroll
    for (int t = 0; t < 8; ++t) acc[t] = v8f{};

    for (int g = 0; g < PROJ_STEPS / 4; ++g) {
#pragma unroll
        for (int u = 0; u < 4; ++u) {
            // A fragment (x^T): M=i (lane%16), K=n
            v2f a;
            a.x = xb[xoff + u * (4 * CIN)];
            a.y = xb[xoff + u * (4 * CIN) + CIN];
#pragma unroll
            for (int t = 0; t < 8; ++t) {
                // B fragment (D): K=n, N=m (lane%16)
                v2f b;
                b.x = Dbp[doff + u * (4 * KM) + t * 16];
                b.y = Dbp[doff + u * (4 * KM) + t * 16 + KM];
                acc[t] = __builtin_amdgcn_wmma_f32_16x16x4_f32(
                    false, a, false, b, (short)0, acc[t], false, false);
            }
        }
        xb  += 16 * CIN;   // scalar pointer bumps (SALU)
        Dbp += 16 * KM;
    }

    // accumulate partial tile into P via hardware f32 atomics
    float* __restrict__ Pbf = P + (size_t)bf * CIN * KM;
#pragma unroll
    for (int t = 0; t < 8; ++t) {
        const int m = (m_group * 8 + t) * 16 + l16;
#pragma unroll
        for (int r = 0; r < 8; ++r) {
            const int i = i_tile * 16 + r + half * 8;  // C/D layout: M = r + 8*half
            unsafeAtomicAdd(&Pbf[i * KM + m], acc[t][r]);
        }
    }
}

// ---------------------------------------------------------------------------
// Kernel 2: St[bf,m,o] = SCALE * sum_i C[o,i,m] * P[bf,i,m]
// Tiny (8.4 MFLOP). Stored *transposed* (m-major) so kernel 3's B fragments
// load coalesced. Lanes walk m -> coalesced reads of C and P.
// ---------------------------------------------------------------------------
__global__ void __launch_bounds__(256) combine_kernel(const float* __restrict__ C,
                                                      const float* __restrict__ P,
                                                      float* __restrict__ St) {
    const int idx = blockIdx.x * 256 + threadIdx.x;  // 8*64*256 = 131072 threads
    const int m  = idx & (KM - 1);
    const int o  = (idx >> 8) & (COUT - 1);
    const int bf = idx >> 14;
    const float* __restrict__ Pbf = P + (size_t)bf * CIN * KM;
    const float* __restrict__ Co  = C + (size_t)o * CIN * KM;
    float s = 0.0f;
#pragma unroll 8
    for (int i = 0; i < CIN; ++i) s += Co[i * KM + m] * Pbf[i * KM + m];
    St[((size_t)bf * KM + m) * COUT + o] = s * SCALE;
}

// ---------------------------------------------------------------------------
// Kernel 3: out[bf,n,o] = sum_m D[bf,n,m] * St[bf,m,o] + bias[o]
// GEMM M=n-rows x N=Cout(64), K=KM(256). Contraction dim m is the *column*
// dim of D, so stage 64 D rows (exactly 64KB) into LDS with an XOR swizzle:
// float4 slot c4 of row r lives at r*256 + ((c4 ^ (r&15))<<2); A-fragment
// ds_load_b64 reads (16 consecutive rows, same m) hit 16 distinct banks.
// OOB rows are clamped, not zeroed: an A row only feeds its own (guarded)
// output row since the reduction runs over m.
// grid = (ceil(N/64)=313, BF), block = 256 (8 waves: 4 n-tiles x 2 o-pairs)
// ---------------------------------------------------------------------------
__device__ __forceinline__ int lds_swz(int row, int f) {  // f = float index in row
    return row * KM + (((f >> 2) ^ (row & 15)) << 2) + (f & 3);
}

__global__ void __launch_bounds__(256) recon_kernel(const float* __restrict__ D,
                                                    const float* __restrict__ St,
                                                    const float* __restrict__ bias,
                                                    float* __restrict__ out) {
    __shared__ float lds[64 * KM];  // 64 KB

    const int nb     = blockIdx.x;
    const int bf     = blockIdx.y;
    const int n_base = nb * 64;
    const float* __restrict__ Db = D + (size_t)bf * NNODE * KM;
    const float* __restrict__ Sb = St + (size_t)bf * KM * COUT;

    // ---- stage 64 rows of D into LDS (coalesced float4, clamp OOB rows) ----
    const int t = threadIdx.x;
#pragma unroll
    for (int s = 0; s < 16; ++s) {          // 16 float4 per thread
        const int j   = t + s * 256;
        const int row = j >> 6;             // 64 float4 per row
        const int c4  = j & 63;
        const int n   = min(n_base + row, NNODE - 1);
        float4 v = ((const float4*)(Db + (size_t)n * KM))[c4];
        *(float4*)&lds[row * KM + ((c4 ^ (row & 15)) << 2)] = v;
    }
    __syncthreads();

    const int lane = t & 31;
    const int wave = __builtin_amdgcn_readfirstlane(t >> 5);
    const int half = lane >> 4, l16 = lane & 15;
    const int n_tile = wave & 3;   // 4 n-tiles of 16 rows
    const int o_pair = wave >> 2;  // 2 pairs x 2 o-tiles cover Cout=64

    const int row = n_tile * 16 + l16;

    // wave-uniform base for St; lane offset covers half/l16; m0 -> immediates
    const float* __restrict__ Sb2 = Sb + o_pair * 32;            // scalar
    const unsigned soff = (unsigned)(half * (2 * COUT) + l16);   // vgpr

    v8f acc0 = v8f{}, acc1 = v8f{};

#pragma unroll
    for (int m0 = 0; m0 < KM; m0 += 4) {
        const int mk = m0 + half * 2;
        // A fragment (D): M=n (lane%16), K=m  — swizzled b64 LDS read
        v2f a = *(const v2f*)&lds[lds_swz(row, mk)];
        // B fragments (St): K=m, N=o — SADDR-form loads, constant immediates
        v2f b0, b1;
        b0.x = Sb2[soff + m0 * COUT];
        b0.y = Sb2[soff + m0 * COUT + COUT];
        b1.x = Sb2[soff + m0 * COUT + 16];
        b1.y = Sb2[soff + m0 * COUT + COUT + 16];
        acc0 = __builtin_amdgcn_wmma_f32_16x16x4_f32(false, a, false, b0,
                                                     (short)0, acc0, false, false);
        acc1 = __builtin_amdgcn_wmma_f32_16x16x4_f32(false, a, false, b1,
                                                     (short)0, acc1, false, false);
    }

    // ---- epilogue: bias + store ----
    const float ba = bias[o_pair * 32 + l16];
    const float bb = bias[o_pair * 32 + 16 + l16];
    float* __restrict__ ob = out + (size_t)bf * NNODE * COUT + o_pair * 32;
#pragma unroll
    for (int r = 0; r < 8; ++r) {
        const int n = n_base + n_tile * 16 + r + half * 8;  // C/D layout
        if (n < NNODE) {
            ob[(size_t)n * COUT + l16]      = acc0[r] + ba;
            ob[(size_t)n * COUT + 16 + l16] = acc1[r] + bb;
        }
    }
}

// ---------------------------------------------------------------------------
extern "C" void kernel_launch(void* const* d_in, const int* in_sizes, int n_in,
                              void* d_out, int out_size, void* d_ws, size_t ws_size,
                              hipStream_t stream) {
    const float* x    = (const float*)d_in[0];  // (2,4,20000,64)
    const float* D    = (const float*)d_in[1];  // (2,4,20000,256)
    const float* C    = (const float*)d_in[2];  // (1,1,64,64,256)
    const float* bias = (const float*)d_in[3];  // (1,1,1,64)
    float* out = (float*)d_out;                 // (2,4,20000,64)

    float* P  = (float*)d_ws;                   // BF*64*256 = 131072 f32
    float* St = P + (size_t)BF * CIN * KM;      // BF*256*64 = 131072 f32

    const int pelems = BF * CIN * KM;
    zero_kernel<<<(pelems + 255) / 256, 256, 0, stream>>>(P, pelems);

    proj_kernel<<<dim3(PROJ_CHUNKS, BF), 256, 0, stream>>>(x, D, P);

    combine_kernel<<<(BF * COUT * KM) / 256, 256, 0, stream>>>(C, P, St);

    recon_kernel<<<dim3((NNODE + 63) / 64, BF), 256, 0, stream>>>(D, St, bias, out);
}